// CUDASeasonalityLstmAutoencoderCount_53730040873361
// MI455X (gfx1250) — compile-verified
//
#include <hip/hip_runtime.h>

typedef __attribute__((ext_vector_type(2))) float v2f;
typedef __attribute__((ext_vector_type(8))) float v8f;

typedef __attribute__((ext_vector_type(4))) unsigned int tdm_v4u;
typedef __attribute__((ext_vector_type(8))) int          tdm_v8i;
typedef __attribute__((ext_vector_type(4))) int          tdm_v4i;

#define B_       32
#define T_       16384
#define F_       64
#define SEAS_    10
#define G_       (4 * SEAS_)   /* 40 gates */
#define RESERVE_ 1000
#define NTILE_   3             /* ceil(40/16) */
#define MTILES_  ((B_ * T_) / 16)   /* 32768 */
#define MT_PER_WAVE      4
#define WAVES_PER_BLOCK  8
#define CHUNK_   128           /* timesteps per TDM chunk (20 KB) */

// ---------------------------------------------------------------------------
// Kernel 1: xg[m, g] = sum_k x[m, k] * Wih[g, k] + bih[g] + bhh[g]
// M = B*T = 524288, N = 40 (3 tiles of 16), K = 64 (16 steps of 4).
// Exact f32 WMMA (V_WMMA_F32_16X16X4_F32): memory-bound, full precision free.
// ---------------------------------------------------------------------------
__global__ __launch_bounds__(256) void xg_gemm_wmma(
    const float* __restrict__ x,     // [B*T, 64] row-major
    const float* __restrict__ Wih,   // [40, 64]  row-major
    const float* __restrict__ bih,   // [40]
    const float* __restrict__ bhh,   // [40]
    float* __restrict__ xg)          // [B*T, 40]
{
  const int lane  = threadIdx.x & 31;
  const int wave  = threadIdx.x >> 5;
  const int ntile = blockIdx.y;
  const int mbase = (blockIdx.x * WAVES_PER_BLOCK + wave) * MT_PER_WAVE;

  const int nloc  = lane & 15;          // A-row / C-col within tile
  const int khalf = (lane >> 4) * 2;    // K sub-offset: 0 (lanes 0-15) or 2
  const int gate  = ntile * 16 + nloc;
  const int gclmp = gate < G_ ? gate : (G_ - 1);   // clamp OOB loads

  // B matrix (4x16 per WMMA): lane n holds (K=khalf, N=n),(K=khalf+1, N=n).
  v2f breg[16];
  #pragma unroll
  for (int kk = 0; kk < 16; ++kk)
    breg[kk] = *(const v2f*)(Wih + (size_t)gclmp * F_ + kk * 4 + khalf);
  const float bias = bih[gclmp] + bhh[gclmp];

  for (int mi = 0; mi < MT_PER_WAVE; ++mi) {
    const int mtile = mbase + mi;
    const float* xrow = x + (size_t)(mtile * 16 + nloc) * F_;
    v8f c = {};
    #pragma unroll
    for (int kk = 0; kk < 16; ++kk) {
      v2f a = *(const v2f*)(xrow + kk * 4 + khalf);
      c = __builtin_amdgcn_wmma_f32_16x16x4_f32(
              false, a, false, breg[kk], (short)0, c, false, false);
    }
    // C layout: VGPR r -> M = r + (lane>>4)*8, N = lane&15
    if (gate < G_) {
      #pragma unroll
      for (int r = 0; r < 8; ++r) {
        const int m = mtile * 16 + r + (lane >> 4) * 8;
        xg[(size_t)m * G_ + gate] = c[r] + bias;
      }
    }
  }
}

// ---------------------------------------------------------------------------
// TDM: async-copy one contiguous chunk (CHUNK_*G_ dwords) from global to LDS.
// D# per CDNA5 ISA: 1-D tile, data_size=4B, count=1,
// type=2 ("image"). All operands wave-uniform -> descriptor lands in SGPRs.
// Tracked with TENSORcnt. 6-arg builtin form (clang-23 / therock headers).
// ---------------------------------------------------------------------------
__device__ __forceinline__ void tdm_load_chunk(const float* gsrc,
                                               unsigned lds_byte_off) {
  const unsigned long long ga = (unsigned long long)(uintptr_t)gsrc;
  const unsigned td0 = CHUNK_ * G_;          // 5120 elements (dwords)

  tdm_v4u g0;
  g0[0] = 1u;                                 // count=1 (valid), user mode
  g0[1] = lds_byte_off;                       // lds_addr (bytes)
  g0[2] = (unsigned)(ga & 0xFFFFFFFFull);     // global_addr[31:0]
  g0[3] = (unsigned)((ga >> 32) & 0x01FFFFFFull) | 0x80000000u; // [56:32]|type=2

  tdm_v8i g1;
  g1[0] = (int)(2u << 16);                    // workgroup_mask=0, data_size=4B
  g1[1] = (int)((td0 & 0xFFFFu) << 16);       // tensor_dim0[15:0]
  g1[2] = (int)((td0 >> 16) & 0xFFFFu);       // tensor_dim0[31:16], dim1=0
  g1[3] = (int)((td0 & 0xFFFFu) << 16);       // tile_dim0 = 5120
  g1[4] = 0;                                  // tile_dim1=0, tile_dim2=0
  g1[5] = (int)td0;                           // tensor_dim0_stride[31:0]
  g1[6] = 0;                                  // stride hi / dim1_stride lo
  g1[7] = 0;

  tdm_v4i gz4 = {0, 0, 0, 0};                 // groups 2/3 unused (<=2D)
  tdm_v8i gz8 = {0, 0, 0, 0, 0, 0, 0, 0};     // extra group (unused)
  __builtin_amdgcn_tensor_load_to_lds(g0, g1, gz4, gz4, gz8, 0 /*cpol*/);
}

// ---------------------------------------------------------------------------
// Kernel 2: serial scans. One block (one wave32) per batch element.
// xg is streamed into LDS by the TDM, double-buffered (TENSORcnt-tracked),
// so the 16384-step serial chain only ever touches LDS.
// Encoder: lane L computes gate L (and gate L+32 for L<8); one v_exp + rcp
// per gate; ds_bpermute regroups (i,f,g,o) per hidden unit & broadcasts h.
// Decoder (hidden=1): scalar scan on lane 0, fused fc + mask + store.
// ---------------------------------------------------------------------------
__device__ __forceinline__ float fast_sigmoid(float a) {
  float e = __expf(-a);
  return 1.0f / (e + 1.0f);
}
__device__ __forceinline__ float fast_tanh(float a) {
  float e = __expf(2.0f * a);
  return (e - 1.0f) / (e + 1.0f);
}

__global__ __launch_bounds__(32) void lstm_recurrence(
    const float* __restrict__ xg,        // [B*T, 40]
    const float* __restrict__ enc_Whh,   // [40, 10]
    const float* __restrict__ dec_Wih,   // [4, 10]
    const float* __restrict__ dec_Whh,   // [4, 1]
    const float* __restrict__ dec_bih,   // [4]
    const float* __restrict__ dec_bhh,   // [4]
    const float* __restrict__ fc_W,      // [1, 1]
    const float* __restrict__ fc_b,      // [1]
    const int*   __restrict__ lens,      // [B]
    float* __restrict__ encH,            // [B, RESERVE, 10] (masked)
    float* __restrict__ y)               // [B, T]
{
  __shared__ float lds_xg[2][CHUNK_ * G_];   // 2 x 20 KB double buffer

  const int b     = blockIdx.x;
  const int lane  = threadIdx.x;             // 0..31
  const int len_b = lens[b];

  // ---- encoder setup ----
  const int g1 = lane;                       // gates 0..31
  const int g2 = lane + 32;                  // gates 32..39 (lanes 0..7)
  float w1[SEAS_], w2[SEAS_];
  #pragma unroll
  for (int j = 0; j < SEAS_; ++j) {
    w1[j] = enc_Whh[g1 * SEAS_ + j];
    w2[j] = (lane < 8) ? enc_Whh[g2 * SEAS_ + j] : 0.0f;
  }
  const bool tnh = (g1 >= 20 && g1 < 30);    // tanh for g-gates 20..29
  const float sS = tnh ?  2.0f : -1.0f;
  const float sA = tnh ?  1.0f :  0.0f;
  const float sB = tnh ? -1.0f :  1.0f;

  float h[SEAS_];
  #pragma unroll
  for (int j = 0; j < SEAS_; ++j) h[j] = 0.0f;
  float cj = 0.0f;

  const float* xgb = xg + (size_t)b * T_ * G_;
  const unsigned lds_off[2] = { (unsigned)(uintptr_t)&lds_xg[0][0],
                                (unsigned)(uintptr_t)&lds_xg[1][0] };

  // prologue: stream chunk 0
  tdm_load_chunk(xgb, lds_off[0]);
  int bufsel = 0;

  for (int tc = 0; tc < T_; tc += CHUNK_) {
    const bool more = (tc + CHUNK_) < T_;
    if (more) {
      asm volatile("s_wait_dscnt 0" ::: "memory");   // done reading buf^1
      tdm_load_chunk(xgb + (size_t)(tc + CHUNK_) * G_, lds_off[bufsel ^ 1]);
      __builtin_amdgcn_s_wait_tensorcnt(1);          // oldest (current) ready
    } else {
      __builtin_amdgcn_s_wait_tensorcnt(0);          // last chunk ready
    }
    asm volatile("" ::: "memory");                   // LDS now holds chunk

    const float* lbuf = &lds_xg[bufsel][0];
    for (int tt = 0; tt < CHUNK_; ++tt) {
      const int t = tc + tt;
      float a1 = lbuf[tt * G_ + g1];                 // ds_load, bank-clean
      float a2 = 0.0f;
      if (lane < 8) a2 = lbuf[tt * G_ + g2];
      #pragma unroll
      for (int j = 0; j < SEAS_; ++j) {
        a1 = __builtin_fmaf(h[j], w1[j], a1);
        a2 = __builtin_fmaf(h[j], w2[j], a2);
      }
      float e1   = __expf(sS * a1);
      float act1 = (sA * e1 + sB) / (e1 + 1.0f);     // sigmoid or tanh
      float e2   = __expf(-a2);
      float act2 = 1.0f / (e2 + 1.0f);               // o-gates 32..39

      const int jj = lane;                           // hidden unit j = lane
      float vi = __shfl(act1, jj,      32);          // i_j
      float vf = __shfl(act1, jj + 10, 32);          // f_j
      float vg = __shfl(act1, jj + 20, 32);          // g_j
      float oa = __shfl(act1, 30 + jj, 32);          // o_j (j<2)
      float ob = __shfl(act2, jj - 2,  32);          // o_j (j>=2)
      float vo = (jj < 2) ? oa : ob;

      cj = __builtin_fmaf(vf, cj, vi * vg);
      float hj = vo * fast_tanh(cj);

      #pragma unroll
      for (int j = 0; j < SEAS_; ++j) h[j] = __shfl(hj, j, 32);

      const int te = t - (T_ - RESERVE_);
      if (te >= 0 && lane < SEAS_) {
        float mask = (t < len_b) ? 1.0f : 0.0f;
        encH[((size_t)b * RESERVE_ + te) * SEAS_ + lane] = hj * mask;
      }
    }
    bufsel ^= 1;
  }

  __syncthreads();

  // ---- decoder + fc (hidden = 1, scalar on lane 0) ----
  if (lane == 0) {
    float dw[4][SEAS_], wh[4], db[4];
    #pragma unroll
    for (int g = 0; g < 4; ++g) {
      #pragma unroll
      for (int j = 0; j < SEAS_; ++j) dw[g][j] = dec_Wih[g * SEAS_ + j];
      wh[g] = dec_Whh[g];
      db[g] = dec_bih[g] + dec_bhh[g];
    }
    const float fw = fc_W[0], fb = fc_b[0];
    float hd = 0.0f, cd = 0.0f;
    for (int t = 0; t < T_; ++t) {
      float ga = db[0], gf = db[1], gg = db[2], go = db[3];
      if (t < RESERVE_) {                  // dec input nonzero only here
        const float* eh = encH + ((size_t)b * RESERVE_ + t) * SEAS_;
        #pragma unroll
        for (int j = 0; j < SEAS_; ++j) {
          float v = eh[j];
          ga = __builtin_fmaf(v, dw[0][j], ga);
          gf = __builtin_fmaf(v, dw[1][j], gf);
          gg = __builtin_fmaf(v, dw[2][j], gg);
          go = __builtin_fmaf(v, dw[3][j], go);
        }
      }
      ga = __builtin_fmaf(hd, wh[0], ga);
      gf = __builtin_fmaf(hd, wh[1], gf);
      gg = __builtin_fmaf(hd, wh[2], gg);
      go = __builtin_fmaf(hd, wh[3], go);

      float iv = fast_sigmoid(ga);
      float fv = fast_sigmoid(gf);
      float gv = fast_tanh(gg);
      float ov = fast_sigmoid(go);
      cd = __builtin_fmaf(fv, cd, iv * gv);
      hd = ov * fast_tanh(cd);

      float mask = (t < len_b) ? 1.0f : 0.0f;
      y[(size_t)b * T_ + t] = __builtin_fmaf(hd * mask, fw, fb);
    }
  }
}

// ---------------------------------------------------------------------------
extern "C" void kernel_launch(void* const* d_in, const int* in_sizes, int n_in,
                              void* d_out, int out_size, void* d_ws, size_t ws_size,
                              hipStream_t stream) {
  const float* to_x = (const float*)d_in[0];
  const float* eWih = (const float*)d_in[1];
  const float* eWhh = (const float*)d_in[2];
  const float* ebih = (const float*)d_in[3];
  const float* ebhh = (const float*)d_in[4];
  const float* dWih = (const float*)d_in[5];
  const float* dWhh = (const float*)d_in[6];
  const float* dbih = (const float*)d_in[7];
  const float* dbhh = (const float*)d_in[8];
  const float* fcW  = (const float*)d_in[9];
  const float* fcb  = (const float*)d_in[10];
  const int*   lens = (const int*)d_in[11];
  float* y = (float*)d_out;

  float* xg   = (float*)d_ws;                                   // 83.9 MB
  float* encH = (float*)((char*)d_ws +
                         (size_t)B_ * T_ * G_ * sizeof(float)); // +1.28 MB

  dim3 grid1(MTILES_ / (WAVES_PER_BLOCK * MT_PER_WAVE), NTILE_, 1);
  xg_gemm_wmma<<<grid1, dim3(256, 1, 1), 0, stream>>>(to_x, eWih, ebih, ebhh, xg);

  lstm_recurrence<<<dim3(B_, 1, 1), dim3(32, 1, 1), 0, stream>>>(
      xg, eWhh, dWih, dWhh, dbih, dbhh, fcW, fcb, lens, encH, y);
}